// DihedralTerm_64656437674513
// MI455X (gfx1250) — compile-verified
//
#include <hip/hip_runtime.h>
#include <math.h>

#define NBLOCKS 1024
#define TPB     256   // 8 wave32 waves per block
#define CHUNK   4     // elements per thread per grid-stride step (b128 loads)

// Branch-free dihedral energy term.
// cos(n*phi - phase) = cos(n*phi)cos(phase) + sin(n*phi)sin(phase),
// with cos(phi)/sin(phi) built directly from the cross products (no acos),
// Chebyshev-style angle addition for n in {0..4} (AMBER periodicity is an
// exact small integer), and hardware v_sin/v_cos for the bounded phase
// in [0, pi).
__device__ __forceinline__ float dihedral_energy(
    const float* __restrict__ coords,
    int i0, int i1, int i2, int i3,
    float f, float per, float ph)
{
    const float p0x = coords[3*i0+0], p0y = coords[3*i0+1], p0z = coords[3*i0+2];
    const float p1x = coords[3*i1+0], p1y = coords[3*i1+1], p1z = coords[3*i1+2];
    const float p2x = coords[3*i2+0], p2y = coords[3*i2+1], p2z = coords[3*i2+2];
    const float p3x = coords[3*i3+0], p3y = coords[3*i3+1], p3z = coords[3*i3+2];

    // OpenMM-style bond vectors
    const float v1x = p0x - p1x, v1y = p0y - p1y, v1z = p0z - p1z;
    const float v2x = p2x - p1x, v2y = p2y - p1y, v2z = p2z - p1z;
    const float v3x = p2x - p3x, v3y = p2y - p3y, v3z = p2z - p3z;

    // cross12 = v1 x v2 ; cross23 = v2 x v3
    const float c12x = v1y * v2z - v1z * v2y;
    const float c12y = v1z * v2x - v1x * v2z;
    const float c12z = v1x * v2y - v1y * v2x;

    const float c23x = v2y * v3z - v2z * v3y;
    const float c23y = v2z * v3x - v2x * v3z;
    const float c23z = v2x * v3y - v2y * v3x;

    const float s12 = c12x * c12x + c12y * c12y + c12z * c12z;
    const float s23 = c23x * c23x + c23y * c23y + c23z * c23z;

    // 1/(max(|c12|,1e-12) * max(|c23|,1e-12)) via hardware v_rsq_f32
    const float r12 = __frsqrt_rn(fmaxf(s12, 1e-24f));
    const float r23 = __frsqrt_rn(fmaxf(s23, 1e-24f));

    const float dotc = c12x * c23x + c12y * c23y + c12z * c23z;
    float c = dotc * r12 * r23;                 // cos(phi)
    c = fminf(1.0f, fmaxf(-1.0f, c));

    // sin(phi) with the torsion sign folded in (phi = sign * acos(c))
    const float sgn = (v1x * c23x + v1y * c23y + v1z * c23z) < 0.0f ? -1.0f : 1.0f;
    const float s = sgn * sqrtf(fmaxf(1.0f - c * c, 0.0f));

    // cos(k*phi), sin(k*phi) for k = 2,3,4 by angle addition (branch-free)
    const float c2 = c * c - s * s;
    const float s2 = 2.0f * c * s;
    const float c3 = c2 * c - s2 * s;
    const float s3 = s2 * c + c2 * s;
    const float c4 = c3 * c - s3 * s;
    const float s4 = s3 * c + c3 * s;

    const int ni = (int)rintf(fabsf(per));      // exact small integer 1..4
    float cn = (ni >= 1) ? c : 1.0f;            // k=0 -> cos=1, sin=0
    float sn = (ni >= 1) ? s : 0.0f;
    cn = (ni == 2) ? c2 : cn;  sn = (ni == 2) ? s2 : sn;
    cn = (ni == 3) ? c3 : cn;  sn = (ni == 3) ? s3 : sn;
    cn = (ni >= 4) ? c4 : cn;  sn = (ni >= 4) ? s4 : sn;

    // cos(n*phi - phase) = cn*cos(phase) + sn*sin(phase); phase in [0, pi)
    // -> hardware v_sin_f32 / v_cos_f32 (bounded domain, no reduction).
    const float sph = __sinf(ph);
    const float cph = __cosf(ph);

    return f * (1.0f + cn * cph + sn * sph);
}

__global__ __launch_bounds__(TPB) void dihedral_partial_kernel(
    const float* __restrict__ coords,
    const int*   __restrict__ ia,
    const int*   __restrict__ ib,
    const int*   __restrict__ ic,
    const int*   __restrict__ id,
    const float* __restrict__ force,
    const float* __restrict__ period,
    const float* __restrict__ phase,
    float*       __restrict__ partial,
    int n)
{
    const int tid     = threadIdx.x;
    const int gid     = blockIdx.x * TPB + tid;
    const int stride  = NBLOCKS * TPB;           // in chunks
    const int nChunks = (n + CHUNK - 1) / CHUNK;
    float acc = 0.0f;

    for (int c = gid; c < nChunks; c += stride) {
        const int base = c * CHUNK;

        // Amortized prefetch of the next grid-stride tile of the streams
        // (lowers to global_prefetch_b8 on gfx1250).
        const int nb = (c + stride) * CHUNK;
        if (nb < n) {
            __builtin_prefetch(&ia[nb],     0, 1);
            __builtin_prefetch(&ib[nb],     0, 1);
            __builtin_prefetch(&ic[nb],     0, 1);
            __builtin_prefetch(&id[nb],     0, 1);
            __builtin_prefetch(&force[nb],  0, 1);
            __builtin_prefetch(&period[nb], 0, 1);
            __builtin_prefetch(&phase[nb],  0, 1);
        }

        if (base + CHUNK <= n) {
            // Vector path: b128 streaming loads, 4 independent chains.
            const int4   vi = ((const int4*)ia)[c];
            const int4   vj = ((const int4*)ib)[c];
            const int4   vk = ((const int4*)ic)[c];
            const int4   vl = ((const int4*)id)[c];
            const float4 vf = ((const float4*)force)[c];
            const float4 vp = ((const float4*)period)[c];
            const float4 vh = ((const float4*)phase)[c];

            acc += dihedral_energy(coords, vi.x, vj.x, vk.x, vl.x, vf.x, vp.x, vh.x);
            acc += dihedral_energy(coords, vi.y, vj.y, vk.y, vl.y, vf.y, vp.y, vh.y);
            acc += dihedral_energy(coords, vi.z, vj.z, vk.z, vl.z, vf.z, vp.z, vh.z);
            acc += dihedral_energy(coords, vi.w, vj.w, vk.w, vl.w, vf.w, vp.w, vh.w);
        } else {
            // Scalar tail.
            for (int e = base; e < n; ++e) {
                acc += dihedral_energy(coords, ia[e], ib[e], ic[e], id[e],
                                       force[e], period[e], phase[e]);
            }
        }
    }

    // Deterministic block tree reduction in LDS.
    __shared__ float red[TPB];
    red[tid] = acc;
    __syncthreads();
#pragma unroll
    for (int s = TPB / 2; s > 0; s >>= 1) {
        if (tid < s) red[tid] += red[tid + s];
        __syncthreads();
    }
    if (tid == 0) partial[blockIdx.x] = red[0];
}

__global__ __launch_bounds__(NBLOCKS) void dihedral_final_kernel(
    const float* __restrict__ partial,
    float*       __restrict__ out)
{
    const int tid = threadIdx.x;
    __shared__ float red[NBLOCKS];
    red[tid] = partial[tid];
    __syncthreads();
#pragma unroll
    for (int s = NBLOCKS / 2; s > 0; s >>= 1) {
        if (tid < s) red[tid] += red[tid + s];
        __syncthreads();
    }
    if (tid == 0) out[0] = red[0];
}

extern "C" void kernel_launch(void* const* d_in, const int* in_sizes, int n_in,
                              void* d_out, int out_size, void* d_ws, size_t ws_size,
                              hipStream_t stream) {
    const float* coords = (const float*)d_in[0];
    const int*   ia     = (const int*)d_in[1];
    const int*   ib     = (const int*)d_in[2];
    const int*   ic     = (const int*)d_in[3];
    const int*   id     = (const int*)d_in[4];
    const float* force  = (const float*)d_in[5];
    const float* period = (const float*)d_in[6];
    const float* phase  = (const float*)d_in[7];

    const int n = in_sizes[1];          // number of dihedrals
    float* partial = (float*)d_ws;      // NBLOCKS floats of scratch
    float* out     = (float*)d_out;

    dihedral_partial_kernel<<<NBLOCKS, TPB, 0, stream>>>(
        coords, ia, ib, ic, id, force, period, phase, partial, n);
    dihedral_final_kernel<<<1, NBLOCKS, 0, stream>>>(partial, out);
}